// MKMMDLoss_25658134626464
// MI455X (gfx1250) — compile-verified
//
#include <hip/hip_runtime.h>
#include <math.h>

typedef __attribute__((ext_vector_type(2))) float v2f;
typedef __attribute__((ext_vector_type(8))) float v8f;
typedef int v4i __attribute__((vector_size(16)));   // matches async-to-LDS builtin pointee

#define BM 128
#define BN 128
#define BK 32
#define LDSS 36   // LDS row stride (floats): 16B-aligned rows + conflict-free frag reads

#if __has_builtin(__builtin_amdgcn_global_load_async_to_lds_b128)
#define ASYNC_COPY 1
#else
#define ASYNC_COPY 0
#endif

__device__ __forceinline__ void wait_async_zero() {
#if __has_builtin(__builtin_amdgcn_s_wait_asynccnt)
    __builtin_amdgcn_s_wait_asynccnt(0);
#else
    asm volatile("s_wait_asynccnt 0x0" ::: "memory");
#endif
}

// ---------------------------------------------------------------- zero accs
__global__ void mmd_zero3(float* acc) {
    if (threadIdx.x < 3) acc[threadIdx.x] = 0.0f;
}

// ---------------------------------------------------------------- row ||x||^2
__global__ __launch_bounds__(256)
void mmd_row_sumsq(const float* __restrict__ X, float* __restrict__ out, int K) {
    const int lane = threadIdx.x & 31;
    const int row  = blockIdx.x * 8 + (threadIdx.x >> 5);
    const float4* xp = (const float4*)(X + (size_t)row * K);
    float s = 0.0f;
    const int nq = K >> 2;
    for (int i = lane; i < nq; i += 32) {
        float4 v = xp[i];
        s += v.x * v.x + v.y * v.y + v.z * v.z + v.w * v.w;
    }
    #pragma unroll
    for (int off = 16; off > 0; off >>= 1) s += __shfl_down(s, off);
    if (lane == 0) out[row] = s;
}

// ---------------------------------------------------------------- fused MMD GEMM
__global__ __launch_bounds__(256)
void mmd_tile(const float* __restrict__ X, const float* __restrict__ Y,
              const float* __restrict__ x2, const float* __restrict__ y2,
              float* __restrict__ acc, int K, float invNM) {
    __shared__ float As[2][BM * LDSS];
    __shared__ float Bs[2][BN * LDSS];

    const int tid   = threadIdx.x;
    const int lane  = tid & 31;
    const int wid   = tid >> 5;        // 0..7
    const int l16   = lane & 15;
    const int khalf = lane >> 4;       // 0 or 1
    const int wm    = (wid & 3) * 32;  // wave M offset within block tile
    const int wn    = (wid >> 2) * 64; // wave N offset within block tile

    const long block_m = (long)blockIdx.y * BM;
    const long block_n = (long)blockIdx.x * BN;

    v8f c[2][4] = {};

    // per-thread chunk mapping for the 128x32 tile copy (4 x float4 per tile)
    int crow[4], ccol[4];
    #pragma unroll
    for (int i = 0; i < 4; ++i) {
        int chunk = tid + i * 256;     // 0..1023
        crow[i] = chunk >> 3;          // 0..127
        ccol[i] = (chunk & 7) << 2;    // 0,4,...,28
    }

    // compute phase on buffer `b`
    auto compute = [&](int b) {
        const float* Ab = &As[b][0];
        const float* Bb = &Bs[b][0];
        #pragma unroll
        for (int kk = 0; kk < BK; kk += 4) {
            v2f a[2], bb[4];
            #pragma unroll
            for (int mt = 0; mt < 2; ++mt)
                a[mt] = *(const v2f*)&Ab[(wm + mt * 16 + l16) * LDSS + kk + 2 * khalf];
            #pragma unroll
            for (int nt = 0; nt < 4; ++nt)
                bb[nt] = *(const v2f*)&Bb[(wn + nt * 16 + l16) * LDSS + kk + 2 * khalf];
            #pragma unroll
            for (int mt = 0; mt < 2; ++mt)
                #pragma unroll
                for (int nt = 0; nt < 4; ++nt)
                    c[mt][nt] = __builtin_amdgcn_wmma_f32_16x16x4_f32(
                        false, a[mt], false, bb[nt], (short)0, c[mt][nt], false, false);
        }
    };

#if ASYNC_COPY
    // ---- CDNA5 path: GLOBAL_LOAD_ASYNC_TO_LDS_B128 (ASYNCcnt), double buffered
    auto issue = [&](int k0, int b) {
        #pragma unroll
        for (int i = 0; i < 4; ++i) {
            __builtin_amdgcn_global_load_async_to_lds_b128(
                (__attribute__((address_space(1))) v4i*)
                    (X + (block_m + crow[i]) * (long)K + k0 + ccol[i]),
                (__attribute__((address_space(3))) v4i*)
                    (&As[b][crow[i] * LDSS + ccol[i]]), 0, 0);
            __builtin_amdgcn_global_load_async_to_lds_b128(
                (__attribute__((address_space(1))) v4i*)
                    (Y + (block_n + crow[i]) * (long)K + k0 + ccol[i]),
                (__attribute__((address_space(3))) v4i*)
                    (&Bs[b][crow[i] * LDSS + ccol[i]]), 0, 0);
        }
    };

    issue(0, 0);
    int cur = 0;
    for (int k0 = 0; k0 < K; k0 += BK) {
        wait_async_zero();          // this wave's copies into buf[cur] done
        __syncthreads();            // all waves' copies visible; all past prev compute
        if (k0 + BK < K) issue(k0 + BK, cur ^ 1);  // overlaps with compute below
        compute(cur);
        cur ^= 1;
    }
#else
    // ---- fallback: register-staged, software pipelined (batched loads, 1 barrier/iter)
    float4 ra[4], rb[4];
    auto loadg = [&](int k0) {
        #pragma unroll
        for (int i = 0; i < 4; ++i) {
            ra[i] = *(const float4*)(X + (block_m + crow[i]) * (long)K + k0 + ccol[i]);
            rb[i] = *(const float4*)(Y + (block_n + crow[i]) * (long)K + k0 + ccol[i]);
        }
    };
    auto storel = [&](int b) {
        #pragma unroll
        for (int i = 0; i < 4; ++i) {
            *(float4*)&As[b][crow[i] * LDSS + ccol[i]] = ra[i];
            *(float4*)&Bs[b][crow[i] * LDSS + ccol[i]] = rb[i];
        }
    };

    loadg(0);
    int cur = 0;
    for (int k0 = 0; k0 < K; k0 += BK) {
        storel(cur);                       // waits only on its own loads
        if (k0 + BK < K) loadg(k0 + BK);   // prefetch next tile, overlaps compute
        __syncthreads();
        compute(cur);
        cur ^= 1;
    }
#endif

    // ---- epilogue: d2 -> multi-bandwidth exp kernel sum
    const float ib0 = 1.0f, ib1 = 0.2f, ib2 = 0.1f,
                ib3 = 0.05f, ib4 = 0.02f, ib5 = 0.01f;
    float part = 0.0f;

    float y2n[4];
    #pragma unroll
    for (int nt = 0; nt < 4; ++nt)
        y2n[nt] = y2[block_n + wn + nt * 16 + l16];

    #pragma unroll
    for (int mt = 0; mt < 2; ++mt) {
        #pragma unroll
        for (int r = 0; r < 8; ++r) {
            // C/D layout: VGPR r -> M = r (lanes 0-15) / r+8 (lanes 16-31)
            float x2m = x2[block_m + wm + mt * 16 + 8 * khalf + r];
            #pragma unroll
            for (int nt = 0; nt < 4; ++nt) {
                float d2 = x2m + y2n[nt] - 2.0f * c[mt][nt][r];
                d2 = fmaxf(d2, 0.0f);
                part += __expf(-d2 * ib0) + __expf(-d2 * ib1) + __expf(-d2 * ib2)
                      + __expf(-d2 * ib3) + __expf(-d2 * ib4) + __expf(-d2 * ib5);
            }
        }
    }

    part *= invNM;
    #pragma unroll
    for (int off = 16; off > 0; off >>= 1) part += __shfl_down(part, off);
    if (lane == 0) atomicAdd(acc, part);
}

// ---------------------------------------------------------------- finalize
__global__ void mmd_finalize(const float* __restrict__ acc, float* __restrict__ out) {
    if (threadIdx.x == 0)
        out[0] = (acc[0] - 2.0f * acc[1] + acc[2]) * (1.0f / 6.0f);
}

// ---------------------------------------------------------------- launcher
extern "C" void kernel_launch(void* const* d_in, const int* in_sizes, int n_in,
                              void* d_out, int out_size, void* d_ws, size_t ws_size,
                              hipStream_t stream) {
    const float* src = (const float*)d_in[0];
    const float* tgt = (const float*)d_in[1];
    const int D = 1024;                    // feature dim fixed by reference
    const int A = in_sizes[0] / D;         // 8192
    const int B = in_sizes[1] / D;         // 8192

    float* ws   = (float*)d_ws;
    float* accv = ws;                      // 3 accumulators (padded to 16)
    float* x2   = ws + 16;                 // A floats
    float* y2   = x2 + A;                  // B floats

    mmd_zero3<<<1, 32, 0, stream>>>(accv);
    mmd_row_sumsq<<<A / 8, 256, 0, stream>>>(src, x2, D);
    mmd_row_sumsq<<<B / 8, 256, 0, stream>>>(tgt, y2, D);

    const float iAA = 1.0f / ((float)A * (float)A);
    const float iAB = 1.0f / ((float)A * (float)B);
    const float iBB = 1.0f / ((float)B * (float)B);

    mmd_tile<<<dim3(A / BN, A / BM), 256, 0, stream>>>(src, src, x2, x2, accv + 0, D, iAA);
    mmd_tile<<<dim3(B / BN, A / BM), 256, 0, stream>>>(src, tgt, x2, y2, accv + 1, D, iAB);
    mmd_tile<<<dim3(B / BN, B / BM), 256, 0, stream>>>(tgt, tgt, y2, y2, accv + 2, D, iBB);

    mmd_finalize<<<1, 32, 0, stream>>>(accv, (float*)d_out);
}